// GaussianAdapter_3DVox_38001870635883
// MI455X (gfx1250) — compile-verified
//
#include <hip/hip_runtime.h>
#include <hip/hip_bf16.h>
#include <cstdint>

// ---------------------------------------------------------------------------
// GaussianAdapter (3D gaussian splat parameter decode), MI455X / gfx1250.
// Memory-bound: ~1.45 GB traffic, ~0.4 GFLOP -> target is HBM roofline
// (~62 us at 23.3 TB/s). No matmul in the workload, so no WMMA; the CDNA5
// feature used is the async global->LDS copy path (ASYNCcnt) to turn the
// 328-byte AoS gaussian records into coalesced b128 streams.
// ---------------------------------------------------------------------------

#define TPB        128   // threads per block (4 waves of 32)
#define GPB        128   // gaussians per block (1 per thread)
#define RAW_STRIDE 82    // floats per raw gaussian record
#define NSH        75    // 3 * 25 SH coefficients

// Block staging geometry: 128*82*4 = 41984 B = 20 full rounds of 128 lanes x
// 16 B (2048 B each) + one 64-lane tail round.
#define FULL_ROUNDS 20
#define ROUND_BYTES 2048
#define TAIL_LANES  64
#define TAIL_OFFSET (FULL_ROUNDS * ROUND_BYTES) // 40960

typedef int v4i __attribute__((vector_size(4 * sizeof(int))));
typedef __attribute__((address_space(1))) v4i gv4i_t;  // global
typedef __attribute__((address_space(3))) v4i lv4i_t;  // LDS

#define HAS_ASYNC_LDS __has_builtin(__builtin_amdgcn_global_load_async_to_lds_b128)

// One 16-byte async copy global->LDS at compile-time immediate offset OFS.
// Per ISA 10.7/§4.4: INST_OFFSET is added to BOTH the global and LDS address.
template <int OFS>
__device__ __forceinline__ void async_b128_at(const float* gaddr, float* laddr) {
#if HAS_ASYNC_LDS
    __builtin_amdgcn_global_load_async_to_lds_b128(
        (gv4i_t*)(uintptr_t)gaddr,
        (lv4i_t*)(uint32_t)(uintptr_t)laddr,
        OFS, /*cpol=*/0);
#else
    uint32_t lds_addr = (uint32_t)(uintptr_t)laddr + OFS;
    asm volatile("global_load_async_to_lds_b128 %0, %1, off offset:%2"
                 :
                 : "v"(lds_addr), "v"(gaddr + OFS / 4), "i"(0)
                 : "memory");
#endif
}

template <int K>
__device__ __forceinline__ void async_stage_rounds(const float* gaddr, float* laddr) {
    async_b128_at<K * ROUND_BYTES>(gaddr, laddr);
    if constexpr (K + 1 < FULL_ROUNDS) async_stage_rounds<K + 1>(gaddr, laddr);
}

__device__ __forceinline__ void wait_asynccnt0() {
#if __has_builtin(__builtin_amdgcn_s_wait_asynccnt)
    __builtin_amdgcn_s_wait_asynccnt(0);
#else
    asm volatile("s_wait_asynccnt 0" ::: "memory");
#endif
}

__device__ __forceinline__ float sh_mask_of(int j) {
    // mask[j]: 1.0 for j==0; 0.1*0.25^d for band d (indices d^2 .. (d+1)^2-1)
    return (j == 0)  ? 1.0f
         : (j < 4)   ? 0.025f        // d=1
         : (j < 9)   ? 0.00625f      // d=2
         : (j < 16)  ? 0.0015625f    // d=3
                     : 0.000390625f; // d=4
}

__global__ __launch_bounds__(TPB)
void gaussian_adapter_kernel(const float* __restrict__ raw,
                             float* __restrict__ out,
                             long long G) {
    __shared__ float smem[GPB * RAW_STRIDE]; // 41,984 bytes
    __shared__ float mask_tab[NSH];          // 300 bytes

    const long long g0 = (long long)blockIdx.x * GPB;
    const int t = threadIdx.x;

    // Output section bases (floats): means(3G) | cov(9G) | scales(3G) |
    // quat(4G) | sh(75G) | opac(G). means/opac handled by memcpyAsync.
    float* __restrict__ out_cov    = out + 3LL  * G;
    float* __restrict__ out_scales = out + 12LL * G;
    float* __restrict__ out_quat   = out + 15LL * G;
    float* __restrict__ out_sh     = out + 19LL * G;

    const bool full = (g0 + GPB) <= G; // uniform per block

    if (full) {
        // ---- Stage 128 x 82 floats into LDS: 21 straight-line async b128 ops
        // sharing one per-lane base address, immediate offsets k*2048.
        const float* gaddr = raw + g0 * RAW_STRIDE + t * 4; // lane base (16B)
        float* laddr = smem + t * 4;
        async_stage_rounds<0>(gaddr, laddr);
        if (t < TAIL_LANES) {
            async_b128_at<TAIL_OFFSET>(gaddr, laddr);
        }
        // SH mask lookup table (overlaps with in-flight DMA).
        if (t < NSH) {
            mask_tab[t] = sh_mask_of(t % 25);
        }
        wait_asynccnt0();
        __syncthreads();
    }

    const long long g = g0 + t;

    // ---- Per-thread decode: scales, quat, cov ----
    if (g < G) {
        float v[7];
        if (full) {
#pragma unroll
            for (int k = 0; k < 7; ++k) v[k] = smem[t * RAW_STRIDE + k];
        } else {
#pragma unroll
            for (int k = 0; k < 7; ++k) v[k] = raw[g * RAW_STRIDE + k];
        }

        // scales = 0.5 + 14.5 * sigmoid(x)
        const float sc0 = 0.5f + 14.5f / (1.0f + expf(-v[0]));
        const float sc1 = 0.5f + 14.5f / (1.0f + expf(-v[1]));
        const float sc2 = 0.5f + 14.5f / (1.0f + expf(-v[2]));

        // quat normalize (norm + 1e-8 in denominator, as in reference)
        const float nrm = sqrtf(v[3]*v[3] + v[4]*v[4] + v[5]*v[5] + v[6]*v[6]);
        const float inv = 1.0f / (nrm + 1e-8f);
        const float qr = v[3] * inv, qi = v[4] * inv, qj = v[5] * inv, qk = v[6] * inv;

        // Rotation matrix
        const float R00 = 1.0f - 2.0f * (qj*qj + qk*qk);
        const float R01 = 2.0f * (qi*qj - qk*qr);
        const float R02 = 2.0f * (qi*qk + qj*qr);
        const float R10 = 2.0f * (qi*qj + qk*qr);
        const float R11 = 1.0f - 2.0f * (qi*qi + qk*qk);
        const float R12 = 2.0f * (qj*qk - qi*qr);
        const float R20 = 2.0f * (qi*qk - qj*qr);
        const float R21 = 2.0f * (qj*qk + qi*qr);
        const float R22 = 1.0f - 2.0f * (qi*qi + qj*qj);

        // M = R * diag(scales); cov = M * M^T (symmetric)
        const float M00 = R00*sc0, M01 = R01*sc1, M02 = R02*sc2;
        const float M10 = R10*sc0, M11 = R11*sc1, M12 = R12*sc2;
        const float M20 = R20*sc0, M21 = R21*sc1, M22 = R22*sc2;

        const float c00 = M00*M00 + M01*M01 + M02*M02;
        const float c01 = M00*M10 + M01*M11 + M02*M12;
        const float c02 = M00*M20 + M01*M21 + M02*M22;
        const float c11 = M10*M10 + M11*M11 + M12*M12;
        const float c12 = M10*M20 + M11*M21 + M12*M22;
        const float c22 = M20*M20 + M21*M21 + M22*M22;

        float* cp = out_cov + g * 9;
        cp[0] = c00; cp[1] = c01; cp[2] = c02;
        cp[3] = c01; cp[4] = c11; cp[5] = c12;
        cp[6] = c02; cp[7] = c12; cp[8] = c22;

        float* sp = out_scales + g * 3;
        sp[0] = sc0; sp[1] = sc1; sp[2] = sc2;

        // quat output is 16B aligned -> single b128 store
        float4 q4; q4.x = qr; q4.y = qi; q4.z = qj; q4.w = qk;
        *(float4*)(out_quat + g * 4) = q4;
    }

    // ---- SH: cooperative, perfectly coalesced dword stores from LDS ----
    if (full) {
        float* __restrict__ shout = out_sh + g0 * NSH;
        const int n = GPB * NSH; // 9600 -> exactly 75 iterations per thread
#pragma unroll 5
        for (int i = t; i < n; i += TPB) {
            const int gl = i / NSH;
            const int r  = i - gl * NSH;
            shout[i] = smem[gl * RAW_STRIDE + 7 + r] * mask_tab[r];
        }
    } else if (g < G) {
        const float* rp = raw + g * RAW_STRIDE + 7;
        float* so = out_sh + g * NSH;
        for (int r = 0; r < NSH; ++r) {
            so[r] = rp[r] * sh_mask_of(r % 25);
        }
    }
}

extern "C" void kernel_launch(void* const* d_in, const int* in_sizes, int n_in,
                              void* d_out, int out_size, void* d_ws, size_t ws_size,
                              hipStream_t stream) {
    const float* means = (const float*)d_in[0]; // (B,N,3)
    const float* opac  = (const float*)d_in[1]; // (B,N)
    const float* raw   = (const float*)d_in[2]; // (B,N,82)
    float* out = (float*)d_out;

    const long long G = (long long)in_sizes[1]; // B*N gaussians

    // Pass-through outputs via DMA copies (graph-capture safe).
    (void)hipMemcpyAsync(out, means, sizeof(float) * 3 * (size_t)G,
                         hipMemcpyDeviceToDevice, stream);
    (void)hipMemcpyAsync(out + 94LL * G, opac, sizeof(float) * (size_t)G,
                         hipMemcpyDeviceToDevice, stream);

    const long long nblocks = (G + GPB - 1) / GPB;
    gaussian_adapter_kernel<<<dim3((unsigned)nblocks), dim3(TPB), 0, stream>>>(raw, out, G);
}